// Model_33947421507685
// MI455X (gfx1250) — compile-verified
//
#include <hip/hip_runtime.h>

typedef __attribute__((ext_vector_type(2))) float v2f;
typedef __attribute__((ext_vector_type(8))) float v8f;

constexpr int B_ = 8192;
constexpr int K_ = 25;
constexpr int L_ = 25;
constexpr float EPS = 1e-5f;

// ---------------- Phase 1: BatchNorm batch statistics ----------------
// One block per column l; writes ws[l] = gamma*rsqrt(var+eps), ws[L+l] = bias - mean*scale.
__global__ void bn_stats_kernel(const float* __restrict__ x,
                                const float* __restrict__ gamma,
                                const float* __restrict__ bnb,
                                float* __restrict__ ws) {
  const int l = blockIdx.x;
  const int tid = threadIdx.x;
  float s = 0.f, s2 = 0.f;
  for (int b = tid; b < B_; b += 256) {
    float v = x[b * L_ + l];
    s += v;
    s2 += v * v;
  }
  __shared__ float ss[256], sq[256];
  ss[tid] = s;
  sq[tid] = s2;
  __syncthreads();
  for (int off = 128; off > 0; off >>= 1) {
    if (tid < off) {
      ss[tid] += ss[tid + off];
      sq[tid] += sq[tid + off];
    }
    __syncthreads();
  }
  if (tid == 0) {
    float mean = ss[0] * (1.f / B_);
    float var = sq[0] * (1.f / B_) - mean * mean;
    float sc = gamma[l] * rsqrtf(var + EPS);
    ws[l] = sc;
    ws[L_ + l] = bnb[l] - mean * sc;
  }
}

// ---------------- Phase 2: fused per-(k,l) MLPs via WMMA ----------------
constexpr int WAVES = 4;
constexpr int BLOCK = WAVES * 32;
constexpr int ACC_STRIDE = K_ + 2;  // 27: gcd(27,64)==1 -> bank-conflict-free

__global__ __launch_bounds__(BLOCK) void mlp_kernel(
    const float* __restrict__ x,
    const float* __restrict__ W1, const float* __restrict__ b1,
    const float* __restrict__ W2, const float* __restrict__ b2,
    const float* __restrict__ W3, const float* __restrict__ b3,
    const float* __restrict__ W4, const float* __restrict__ b4,
    const float* __restrict__ alpha, const float* __restrict__ beta,
    const float* __restrict__ ws, float* __restrict__ out) {
  __shared__ float sW1[K_][8], sB1[K_][8];
  __shared__ float sW2[K_][64], sB2[K_][8];
  __shared__ float sW3[K_][48], sB3[K_][6];
  __shared__ float sW4[K_][6], sB4[K_];
  __shared__ float sAlpha[L_ * K_];
  __shared__ float sBeta[K_];
  __shared__ float sScale[L_], sShift[L_];
  __shared__ float sAcc[WAVES][32][ACC_STRIDE];  // per-thread private rows

  const int tid = threadIdx.x;
  const unsigned lane = tid & 31u;
  const int wave = tid >> 5;
  const int m = (int)(lane & 15u);    // batch row within 16-row tile
  const int half = (int)(lane >> 4);  // lane half selects K-subrows of fragment
  const int rowbase = (blockIdx.x * WAVES + wave) * 16;

  for (int t = tid; t < L_ * K_; t += BLOCK) sAlpha[t] = alpha[t];
  for (int t = tid; t < K_; t += BLOCK) sBeta[t] = beta[t];
  for (int t = tid; t < L_; t += BLOCK) {
    sScale[t] = ws[t];
    sShift[t] = ws[L_ + t];
  }
  // Each thread owns its sAcc row exclusively (no barrier needed for it).
  float* myAcc = &sAcc[wave][lane][0];
  for (int k = 0; k < K_; ++k) myAcc[k] = 0.f;

  for (int l = 0; l < L_; ++l) {
    __syncthreads();
    // Stage all-k weight slices for this l into LDS with 128-bit copies.
    for (int t = tid; t < K_ * 2; t += BLOCK) {  // W1/b1/b2: 8 floats = 2x float4
      int k = t >> 1, q = t & 1;
      ((float4*)sW1[k])[q] = ((const float4*)(W1 + (k * L_ + l) * 8))[q];
      ((float4*)sB1[k])[q] = ((const float4*)(b1 + (k * L_ + l) * 8))[q];
      ((float4*)sB2[k])[q] = ((const float4*)(b2 + (k * L_ + l) * 8))[q];
    }
    for (int t = tid; t < K_ * 16; t += BLOCK) {  // W2: 64 floats = 16x float4
      int k = t >> 4, q = t & 15;
      ((float4*)sW2[k])[q] = ((const float4*)(W2 + (k * L_ + l) * 64))[q];
    }
    for (int t = tid; t < K_ * 12; t += BLOCK) {  // W3: 48 floats = 12x float4
      int k = t / 12, q = t % 12;
      ((float4*)sW3[k])[q] = ((const float4*)(W3 + (k * L_ + l) * 48))[q];
    }
    for (int t = tid; t < K_ * 6; t += BLOCK) {
      int k = t / 6, j = t % 6;
      sB3[k][j] = b3[(k * L_ + l) * 6 + j];
      sW4[k][j] = W4[(k * L_ + l) * 6 + j];
    }
    for (int t = tid; t < K_; t += BLOCK) sB4[t] = b4[t * L_ + l];
    __syncthreads();

    const float xv = x[(rowbase + m) * L_ + l] * sScale[l] + sShift[l];
    const int h0 = half * 2;      // K-chunk 0 covers h = {h0, h0+1}
    const int h1 = 4 + half * 2;  // K-chunk 1 covers h = {h1, h1+1}

#pragma unroll 2
    for (int k = 0; k < K_; ++k) {
      // B fragments = layer-1 activations h1[h, b] (lane-local rank-1 + relu).
      v2f bf0, bf1;
      bf0.x = fmaxf(fmaf(xv, sW1[k][h0], sB1[k][h0]), 0.f);
      bf0.y = fmaxf(fmaf(xv, sW1[k][h0 + 1], sB1[k][h0 + 1]), 0.f);
      bf1.x = fmaxf(fmaf(xv, sW1[k][h1], sB1[k][h1]), 0.f);
      bf1.y = fmaxf(fmaf(xv, sW1[k][h1 + 1], sB1[k][h1 + 1]), 0.f);
      // A fragments = W2 rows (i = m); rows i>=8 zero-padded via selects.
      const bool act = (m < 8);
      v2f af0, af1;
      af0.x = act ? sW2[k][m * 8 + h0] : 0.f;
      af0.y = act ? sW2[k][m * 8 + h0 + 1] : 0.f;
      af1.x = act ? sW2[k][m * 8 + h1] : 0.f;
      af1.y = act ? sW2[k][m * 8 + h1 + 1] : 0.f;
      // C init = b2 broadcast on M=0..7 rows; M=8..15 rows are zero.
      v8f c;
#pragma unroll
      for (int r = 0; r < 8; ++r) c[r] = (half == 0) ? sB2[k][r] : 0.f;
      // D[i,b] = sum_h W2[i,h] * h1[h,b] + b2[i]  (two K=4 chunks)
      c = __builtin_amdgcn_wmma_f32_16x16x4_f32(false, af0, false, bf0,
                                                (short)0, c, false, false);
      c = __builtin_amdgcn_wmma_f32_16x16x4_f32(false, af1, false, bf1,
                                                (short)0, c, false, false);
      // Lanes 0..15 now hold the complete h2[0..7] for batch row m.
      float h2[8];
#pragma unroll
      for (int r = 0; r < 8; ++r) h2[r] = fmaxf(c[r], 0.f);
      float f = sB4[k];
#pragma unroll
      for (int j = 0; j < 6; ++j) {
        float t = sB3[k][j];
#pragma unroll
        for (int i = 0; i < 8; ++i) t = fmaf(sW3[k][j * 8 + i], h2[i], t);
        f = fmaf(fmaxf(t, 0.f), sW4[k][j], f);
      }
      // Accumulate over l in this thread's private LDS row (dynamic k index
      // without forcing full unroll / register blowup). Lanes 16..31 write
      // their own junk rows which are never read back.
      myAcc[k] = fmaf(f, sAlpha[l * K_ + k], myAcc[k]);
    }
  }

  if (lane < 16u) {
    for (int k = 0; k < K_; ++k)
      out[(rowbase + (int)lane) * K_ + k] = sBeta[k] + myAcc[k];
  }
}

extern "C" void kernel_launch(void* const* d_in, const int* in_sizes, int n_in,
                              void* d_out, int out_size, void* d_ws, size_t ws_size,
                              hipStream_t stream) {
  const float* x = (const float*)d_in[0];
  const float* gamma = (const float*)d_in[1];
  const float* bnb = (const float*)d_in[2];
  const float* W1 = (const float*)d_in[3];
  const float* b1 = (const float*)d_in[4];
  const float* W2 = (const float*)d_in[5];
  const float* b2 = (const float*)d_in[6];
  const float* W3 = (const float*)d_in[7];
  const float* b3 = (const float*)d_in[8];
  const float* W4 = (const float*)d_in[9];
  const float* b4 = (const float*)d_in[10];
  const float* alpha = (const float*)d_in[11];
  const float* beta = (const float*)d_in[12];
  float* out = (float*)d_out;
  float* ws = (float*)d_ws;

  bn_stats_kernel<<<L_, 256, 0, stream>>>(x, gamma, bnb, ws);

  const int tiles = B_ / 16;  // 512 batch tiles, 4 per block
  mlp_kernel<<<tiles / WAVES, BLOCK, 0, stream>>>(
      x, W1, b1, W2, b2, W3, b3, W4, b4, alpha, beta, ws, out);
}